// DynamicRouting_23862838297055
// MI455X (gfx1250) — compile-verified
//
#include <hip/hip_runtime.h>
#include <hip/hip_bf16.h>

// Shapes (fixed by the reference):
//   u:    [B=128,  N_in=1152, D_in=8]   fp32
//   W:    [N_in=1152, N_out=10, D_out=16, D_in=8] fp32
//   out v:[B=128,  N_out=10, D_out=16]  fp32
#define B_SZ   128
#define NIN    1152
#define NOUT   10
#define DOUT   16
#define DIN    8
#define ODIM   (NOUT * DOUT)   // 160

typedef __attribute__((ext_vector_type(2))) float v2f;
typedef __attribute__((ext_vector_type(4))) float f32x4;
typedef __attribute__((ext_vector_type(8))) float v8f;

__device__ __forceinline__ float clip10(float x) {
    return fminf(fmaxf(x, -10.0f), 10.0f);
}

// ---------------------------------------------------------------------------
// Kernel 1: u_hat[b,i,o,d] = clip( sum_k W[i,o,d,k] * u[b,i,k], +-10 )
// Per i: GEMM  A[160(od) x 8(k)] * B[8(k) x 128(b)] -> D[160 x 128]
// One block per i, 8 waves; wave w = batch tile (16 cols), loop 10 row tiles.
// Uses V_WMMA_F32_16X16X4_F32 (fp32 exact), two chained WMMAs cover K=8.
// ---------------------------------------------------------------------------
__global__ __launch_bounds__(256) void uhat_wmma_kernel(
    const float* __restrict__ u,    // [B, NIN, DIN]
    const float* __restrict__ W,    // [NIN, NOUT, DOUT, DIN]
    float* __restrict__ uhat)       // [B, NIN, ODIM]
{
    const int i    = blockIdx.x;          // input capsule index 0..1151
    const int lane = threadIdx.x & 31;
    const int wave = threadIdx.x >> 5;    // tile_n: 0..7 (batch tile)
    const int half = lane >> 4;           // 0: K-lo lanes, 1: K-hi lanes
    const int l15  = lane & 15;

    // ---- B operand: u[:, i, :]  (4x16 tile per WMMA), independent of tile_m.
    // Layout assumption (symmetric to documented A layout):
    //   lane n = l15 holds column N=n; VGPR0 = K(k0), VGPR1 = K(k0+1),
    //   k0 = 0 for lanes 0-15, 2 for lanes 16-31 (first WMMA), +4 (second).
    const int bcol = wave * 16 + l15;     // batch column 0..127
    const float* up = u + ((size_t)bcol * NIN + i) * DIN;
    const int k0 = half * 2;
    v2f b0, b1;
    b0.x = up[k0];      b0.y = up[k0 + 1];
    b1.x = up[k0 + 4];  b1.y = up[k0 + 5];

    const float* Wi     = W    + (size_t)i * ODIM * DIN;
    float*       uh_col = uhat + ((size_t)bcol * NIN + i) * ODIM;

    #pragma unroll
    for (int tm = 0; tm < 10; ++tm) {
        // ---- A operand: W[i, od-tile, k] (16x4 tile). ISA layout:
        // lane m = l15 holds row M=m; VGPR0=K(k0), VGPR1=K(k0+1),
        // lanes 16-31 hold K2/K3 (and K6/K7 for the second WMMA).
        const int od = tm * 16 + l15;
        const float* wp = Wi + (size_t)od * DIN;
        v2f a0, a1;
        a0.x = wp[k0];      a0.y = wp[k0 + 1];
        a1.x = wp[k0 + 4];  a1.y = wp[k0 + 5];

        v8f acc = {};
        // 8 args: (neg_a, A, neg_b, B, c_mod, C, reuse_a, reuse_b)
        acc = __builtin_amdgcn_wmma_f32_16x16x4_f32(
                false, a0, false, b0, (short)0, acc, false, false);
        acc = __builtin_amdgcn_wmma_f32_16x16x4_f32(
                false, a1, false, b1, (short)0, acc, false, false);

        // D layout: VGPR j, lanes 0-15 -> M=j, lanes 16-31 -> M=j+8; N=l15.
        // => this lane owns 8 consecutive od rows at fixed column bcol.
        const int od_base = tm * 16 + half * 8;
        float* dst = uh_col + od_base;      // 32B aligned
        f32x4 lo, hi;
        lo.x = clip10(acc[0]); lo.y = clip10(acc[1]);
        lo.z = clip10(acc[2]); lo.w = clip10(acc[3]);
        hi.x = clip10(acc[4]); hi.y = clip10(acc[5]);
        hi.z = clip10(acc[6]); hi.w = clip10(acc[7]);
        *(f32x4*)(dst)     = lo;
        *(f32x4*)(dst + 4) = hi;
    }
}

// ---------------------------------------------------------------------------
// Kernel 2: full 3-iteration dynamic routing, one block per batch element.
// 512 threads = 32 groups x 16 lanes (lane = d). Group g owns i = g, g+32, ...
// u_hat[b] (737 KB) is re-read from L2 once per iteration.
// ---------------------------------------------------------------------------
__global__ __launch_bounds__(512) void routing_kernel(
    const float* __restrict__ uhat,   // [B, NIN, ODIM]
    float* __restrict__ out)          // [B, ODIM]
{
    const int b      = blockIdx.x;
    const int tid    = (int)threadIdx.x;
    const int lane   = tid & 31;
    const int waveId = tid >> 5;      // 0..15
    const int lane_d = tid & 15;      // d index
    const int grp    = tid >> 4;      // 0..31
    const int NGRP   = 32;

    __shared__ float blog[NIN * NOUT];       // routing logits b[i,o]  (46080 B)
    __shared__ float v_sh[ODIM];             // current v[o,d]         (640 B)
    __shared__ float s_red[16 * ODIM];       // per-wave s partials    (10240 B)
    __shared__ float sq_sh[NOUT];            // squared norms per o

    for (int idx = tid; idx < NIN * NOUT; idx += blockDim.x) blog[idx] = 0.0f;
    __syncthreads();

    const float* uh_base = uhat + (size_t)b * NIN * ODIM;

    for (int it = 0; it < 3; ++it) {
        float s_acc[NOUT];
        #pragma unroll
        for (int o = 0; o < NOUT; ++o) s_acc[o] = 0.0f;

        for (int i = grp; i < NIN; i += NGRP) {
            const float* uh_i = uh_base + (size_t)i * ODIM;
            float uh[NOUT];
            #pragma unroll
            for (int o = 0; o < NOUT; ++o) uh[o] = uh_i[o * DOUT + lane_d];

            float c[NOUT];
            if (it == 0) {
                #pragma unroll
                for (int o = 0; o < NOUT; ++o) c[o] = 0.1f;   // softmax(0)
            } else {
                float bnew[NOUT];
                float m = -1e30f;
                #pragma unroll
                for (int o = 0; o < NOUT; ++o) {
                    // agreement[o] = sum_d u_hat[i,o,d] * v[o,d]
                    float p = uh[o] * v_sh[o * DOUT + lane_d];
                    p += __shfl_xor(p, 1, 16);
                    p += __shfl_xor(p, 2, 16);
                    p += __shfl_xor(p, 4, 16);
                    p += __shfl_xor(p, 8, 16);
                    float bn = blog[i * NOUT + o] + clip10(p);
                    bnew[o] = bn;
                    if (lane_d == o) blog[i * NOUT + o] = bn;  // update logits
                    m = fmaxf(m, bn);
                }
                float sum = 0.0f;
                #pragma unroll
                for (int o = 0; o < NOUT; ++o) { c[o] = expf(bnew[o] - m); sum += c[o]; }
                const float inv = 1.0f / sum;
                #pragma unroll
                for (int o = 0; o < NOUT; ++o) c[o] *= inv;
            }
            #pragma unroll
            for (int o = 0; o < NOUT; ++o) s_acc[o] += c[o] * uh[o];
        }

        // reduce the two 16-groups of each wave, then across 16 waves via LDS
        #pragma unroll
        for (int o = 0; o < NOUT; ++o) {
            float p = s_acc[o];
            p += __shfl_xor(p, 16, 32);
            if (lane < 16) s_red[waveId * ODIM + o * DOUT + lane] = p;
        }
        __syncthreads();

        if (tid < ODIM) {
            float s = 0.0f;
            #pragma unroll
            for (int w = 0; w < 16; ++w) s += s_red[w * ODIM + tid];
            v_sh[tid] = s;                      // raw s[o,d]
        }
        __syncthreads();

        // squash: per o, sq = clip(sum_d s^2, 1e-8, 1e4); v = sq/(1+sq) * s/(sqrt(sq)+1e-8)
        if (tid < NOUT) {
            float sq = 0.0f;
            #pragma unroll
            for (int d = 0; d < DOUT; ++d) {
                float t = v_sh[tid * DOUT + d];
                sq += t * t;
            }
            sq_sh[tid] = fminf(fmaxf(sq, 1e-8f), 1e4f);
        }
        __syncthreads();

        if (tid < ODIM) {
            const int o = tid >> 4;
            const float sq    = sq_sh[o];
            const float norm  = sqrtf(sq);
            const float scale = sq / (1.0f + sq);
            v_sh[tid] = scale * (v_sh[tid] / (norm + 1e-8f));
        }
        __syncthreads();
    }

    if (tid < ODIM) out[(size_t)b * ODIM + tid] = v_sh[tid];
}

// ---------------------------------------------------------------------------
extern "C" void kernel_launch(void* const* d_in, const int* in_sizes, int n_in,
                              void* d_out, int out_size, void* d_ws, size_t ws_size,
                              hipStream_t stream) {
    const float* u = (const float*)d_in[0];   // [128, 1152, 8]
    const float* W = (const float*)d_in[1];   // [1152, 10, 16, 8]
    float* out  = (float*)d_out;              // [128, 10, 16]
    float* uhat = (float*)d_ws;               // needs 128*1152*160*4 = 94,371,840 B

    uhat_wmma_kernel<<<NIN, 256, 0, stream>>>(u, W, uhat);
    routing_kernel<<<B_SZ, 512, 0, stream>>>(uhat, out);
}